// EdgeEarlyInteractionBaseline_36000415875807
// MI455X (gfx1250) — compile-verified
//
#include <hip/hip_runtime.h>
#include <math.h>

// ---------------------------------------------------------------------------
// CDNA5 (gfx1250) implementation of the EdgeEarlyInteraction forward pass.
// Dense algebra runs on V_WMMA_F32_16X16X4_F32 (fp32 matrix core, full
// precision vs the f32 reference; workload is L2-bandwidth bound on MI455X).
// GEMM inner loop is branch-free: row/col indices are clamped (garbage flows
// only into never-stored accumulator slots), K-tail handled by selects.
// ---------------------------------------------------------------------------

typedef float v2f __attribute__((ext_vector_type(2)));
typedef float v8f __attribute__((ext_vector_type(8)));

#define WPB 4  // waves per block (4 x wave32 = 128 threads)

__device__ __forceinline__ v8f wmma4(v2f a, v2f w, v8f c) {
  return __builtin_amdgcn_wmma_f32_16x16x4_f32(false, a, false, w, (short)0, c,
                                               false, false);
}

// C[b][M,N] = act( op(A[b]) @ op(W[b]) * alpha + bias ) (+= existing C if ACC)
// A: row-major (lda) or TA (element A[k*lda+m])
// W: row-major KxN (ldw) or TB (element W[n*ldw+k], i.e. B given NxK)
// Each wave computes a 16x32 output tile (one A frag feeds two WMMAs).
template <bool RELU, bool ACC, bool TA, bool TB>
__launch_bounds__(128) __global__
void gemm_wmma_f32(const float* __restrict__ A, const float* __restrict__ W,
                   const float* __restrict__ bias, float* __restrict__ C,
                   int M, int N, int K, int lda, int ldw, int ldc,
                   long sA, long sW, long sC, float alpha) {
  const int b = blockIdx.y;
  A += (long)b * sA;
  W += (long)b * sW;
  C += (long)b * sC;
  const int lane = threadIdx.x & 31;
  const int wid  = threadIdx.x >> 5;
  const int tilesM  = (M + 15) >> 4;
  const int tilesN2 = (N + 31) >> 5;           // 32-col super-tiles
  const int tile = blockIdx.x * WPB + wid;     // wave-uniform -> EXEC all ones
  if (tile >= tilesM * tilesN2) return;
  const int tm  = tile / tilesN2;
  const int tn2 = tile - tm * tilesN2;
  const int l16 = lane & 15;
  const bool hi = lane >= 16;
  const int ko  = hi ? 2 : 0;

  // clamped indices: OOB A-rows / B-cols feed only never-stored C slots
  const int mc  = (tm * 16 + l16 < M) ? tm * 16 + l16 : M - 1;
  const int nA  = tn2 * 32 + l16;
  const int n0c = (nA < N) ? nA : N - 1;
  const int n1c = (nA + 16 < N) ? nA + 16 : N - 1;

  const float* pa  = TA ? (A + mc) : (A + (long)mc * lda);
  const float* pw0 = TB ? (W + (long)n0c * ldw) : (W + n0c);
  const float* pw1 = TB ? (W + (long)n1c * ldw) : (W + n1c);

  auto loadA = [&](int k0, int k1) {
    v2f a;
    if (TA) { a.x = pa[k0 * lda]; a.y = pa[k1 * lda]; }
    else    { a.x = pa[k0];       a.y = pa[k1]; }
    return a;
  };
  auto loadW = [&](const float* pw, int k0, int k1) {
    v2f w;
    if (TB) { w.x = pw[k0];       w.y = pw[k1]; }
    else    { w.x = pw[k0 * ldw]; w.y = pw[k1 * ldw]; }
    return w;
  };

  v8f accs[2] = {v8f{}, v8f{}};
  const int kFull = K & ~3;

  if (kFull > 0) {  // branch-free pipelined main loop
    v2f a  = loadA(ko, ko + 1);
    v2f w0 = loadW(pw0, ko, ko + 1);
    v2f w1 = loadW(pw1, ko, ko + 1);
    for (int kb = 4; kb < kFull; kb += 4) {
      const int k0 = kb + ko, k1 = k0 + 1;
      v2f an  = loadA(k0, k1);
      v2f w0n = loadW(pw0, k0, k1);
      v2f w1n = loadW(pw1, k0, k1);
      accs[0] = wmma4(a, w0, accs[0]);
      accs[1] = wmma4(a, w1, accs[1]);
      a = an; w0 = w0n; w1 = w1n;
    }
    accs[0] = wmma4(a, w0, accs[0]);
    accs[1] = wmma4(a, w1, accs[1]);
  }
  if (kFull < K) {  // K tail: zero A frag via selects (kills k>=K products)
    const int k0 = kFull + ko, k1 = k0 + 1;
    const int k0c = (k0 < K) ? k0 : K - 1;
    const int k1c = (k1 < K) ? k1 : K - 1;
    v2f a  = loadA(k0c, k1c);
    a.x = (k0 < K) ? a.x : 0.f;
    a.y = (k1 < K) ? a.y : 0.f;
    v2f w0 = loadW(pw0, k0c, k1c);
    v2f w1 = loadW(pw1, k0c, k1c);
    accs[0] = wmma4(a, w0, accs[0]);
    accs[1] = wmma4(a, w1, accs[1]);
  }

  // epilogue
#pragma unroll
  for (int sub = 0; sub < 2; ++sub) {
    const int col = tn2 * 32 + sub * 16 + l16;
    if (col < N) {
      const float bv = bias ? bias[col] : 0.f;
#pragma unroll
      for (int v = 0; v < 8; ++v) {
        const int row = tm * 16 + v + (hi ? 8 : 0);
        if (row < M) {
          float val = accs[sub][v] * alpha + bv;
          if (RELU) val = fmaxf(val, 0.f);
          const long idx = (long)row * ldc + col;
          if (ACC) val += C[idx];
          C[idx] = val;
        }
      }
    }
  }
}

// ------------------------- elementwise / helper kernels ---------------------

__global__ void zero_f32(float* __restrict__ p, long n) {
  long i = (long)blockIdx.x * blockDim.x + threadIdx.x;
  if (i < n) p[i] = 0.f;
}

// X1[e][0:32]=e_enc[e], [32:160]=store[e]
__global__ void pack_comb(const float* __restrict__ e_enc,
                          const float* __restrict__ store,
                          float* __restrict__ X, long total) {
  long i = (long)blockIdx.x * blockDim.x + threadIdx.x;
  if (i >= total) return;
  long e = i / 160;
  int c = (int)(i - e * 160);
  X[i] = (c < 32) ? e_enc[e * 32 + c] : store[e * 128 + (c - 32)];
}

// Xm = [h[from], h[to], comb, 0,0] ; Xr = [h[to], h[from], comb, 0,0]
__global__ void pack_msg(const float* __restrict__ h,
                         const float* __restrict__ comb,
                         const int* __restrict__ from_idx,
                         const int* __restrict__ to_idx,
                         float* __restrict__ Xm, float* __restrict__ Xr,
                         long total) {
  long i = (long)blockIdx.x * blockDim.x + threadIdx.x;
  if (i >= total) return;
  long e = i / 160;
  int c = (int)(i - e * 160);
  const int f = from_idx[e], t = to_idx[e];
  float vm, vr;
  if (c < 64) {
    vm = h[(long)f * 64 + c];
    vr = h[(long)t * 64 + c];
  } else if (c < 128) {
    vm = h[(long)t * 64 + (c - 64)];
    vr = h[(long)f * 64 + (c - 64)];
  } else if (c < 158) {
    vm = vr = comb[e * 32 + (c - 128)];
  } else {
    vm = vr = 0.f;
  }
  Xm[i] = vm;
  Xr[i] = vr;
}

// Xu = [agg(128), h(64)]
__global__ void pack_upd(const float* __restrict__ agg,
                         const float* __restrict__ h, float* __restrict__ X,
                         long total) {
  long i = (long)blockIdx.x * blockDim.x + threadIdx.x;
  if (i >= total) return;
  long r = i / 192;
  int c = (int)(i - r * 192);
  X[i] = (c < 128) ? agg[r * 128 + c] : h[r * 64 + (c - 128)];
}

// agg[to[e]] += Mf[e]; agg[from[e]] += Mb[e]   (128 cols)
__global__ void scatter_add_agg(const float* __restrict__ Mf,
                                const float* __restrict__ Mb,
                                const int* __restrict__ from_idx,
                                const int* __restrict__ to_idx,
                                float* __restrict__ agg, long total) {
  long i = (long)blockIdx.x * blockDim.x + threadIdx.x;
  if (i >= total) return;
  long e = i >> 7;
  int c = (int)(i & 127);
  atomicAdd(&agg[(long)to_idx[e] * 128 + c], Mf[i]);
  atomicAdd(&agg[(long)from_idx[e] * 128 + c], Mb[i]);
}

// padded[edge_indices[e]][c] = msgs[e][c]
__global__ void scatter_msgs(const float* __restrict__ msgs,
                             const int* __restrict__ eidx,
                             float* __restrict__ padded, long total) {
  long i = (long)blockIdx.x * blockDim.x + threadIdx.x;
  if (i >= total) return;
  long e = i >> 7;
  int c = (int)(i & 127);
  padded[(long)eidx[e] * 128 + c] = msgs[i];
}

// store[e][c] = inter[edge_indices[e]][c]
__global__ void gather_store(const float* __restrict__ inter,
                             const int* __restrict__ eidx,
                             float* __restrict__ store, long total) {
  long i = (long)blockIdx.x * blockDim.x + threadIdx.x;
  if (i >= total) return;
  long e = i >> 7;
  int c = (int)(i & 127);
  store[i] = inter[(long)eidx[e] * 128 + c];
}

// zero rows m >= counts[g] of T[g*256+m][0:64]
__global__ void mask_rows(float* __restrict__ T, const int* __restrict__ counts,
                          long total) {
  long i = (long)blockIdx.x * blockDim.x + threadIdx.x;
  if (i >= total) return;
  long r = i >> 6;
  int m = (int)(r & 255);
  int g = (int)(r >> 8);
  if (m >= counts[g]) T[i] = 0.f;
}

// row logsumexp subtraction: one block per (b,m), 256 threads over n
__launch_bounds__(256) __global__ void sinkhorn_row(float* __restrict__ S) {
  __shared__ float red[256];
  const long base = (long)blockIdx.x * 256;
  const int n = threadIdx.x;
  const float x = S[base + n];
  red[n] = x;
  __syncthreads();
  for (int s = 128; s > 0; s >>= 1) {
    if (n < s) red[n] = fmaxf(red[n], red[n + s]);
    __syncthreads();
  }
  const float mx = red[0];
  __syncthreads();
  red[n] = __expf(x - mx);
  __syncthreads();
  for (int s = 128; s > 0; s >>= 1) {
    if (n < s) red[n] += red[n + s];
    __syncthreads();
  }
  S[base + n] = x - (mx + __logf(red[0]));
}

// column logsumexp subtraction: one block per pair b, thread = column n
__launch_bounds__(256) __global__ void sinkhorn_col(float* __restrict__ S) {
  float* P = S + (long)blockIdx.x * 256 * 256;
  const int n = threadIdx.x;
  float mx = -INFINITY;
  for (int r = 0; r < 256; ++r) mx = fmaxf(mx, P[r * 256 + n]);
  float s = 0.f;
  for (int r = 0; r < 256; ++r) s += __expf(P[r * 256 + n] - mx);
  const float lse = mx + __logf(s);
  for (int r = 0; r < 256; ++r) P[r * 256 + n] -= lse;
}

__global__ void exp_inplace(float* __restrict__ p, long n) {
  long i = (long)blockIdx.x * blockDim.x + threadIdx.x;
  if (i < n) p[i] = __expf(p[i]);
}

// out[b] = -sum relu(q_store - q_from_c) over 256x128
__launch_bounds__(256) __global__
void final_scores(const float* __restrict__ padded,
                  const float* __restrict__ inter, float* __restrict__ out) {
  __shared__ float red[256];
  const int b = blockIdx.x;
  const float* q  = padded + (long)(2 * b) * 256 * 128;
  const float* qc = inter  + (long)(2 * b) * 256 * 128;
  float s = 0.f;
  for (int i = threadIdx.x; i < 256 * 128; i += 256)
    s += fmaxf(q[i] - qc[i], 0.f);
  red[threadIdx.x] = s;
  __syncthreads();
  for (int st = 128; st > 0; st >>= 1) {
    if (threadIdx.x < st) red[threadIdx.x] += red[threadIdx.x + st];
    __syncthreads();
  }
  if (threadIdx.x == 0) out[b] = -red[0];
}

// ------------------------------ host helpers -------------------------------

static inline int nblk(long n, int t = 256) { return (int)((n + t - 1) / t); }

enum GMode { NN = 0, NT = 1, TN = 2 };

static void gemm(hipStream_t st, GMode mode, bool relu, bool accum,
                 const float* A, const float* W, const float* bias, float* C,
                 int M, int N, int K, int lda, int ldw, int ldc,
                 int batch = 1, long sA = 0, long sW = 0, long sC = 0,
                 float alpha = 1.f) {
  const int tiles = ((M + 15) / 16) * ((N + 31) / 32);
  dim3 grid((tiles + WPB - 1) / WPB, batch), block(WPB * 32);
  if (mode == NN && !relu && !accum)
    gemm_wmma_f32<false, false, false, false><<<grid, block, 0, st>>>(
        A, W, bias, C, M, N, K, lda, ldw, ldc, sA, sW, sC, alpha);
  else if (mode == NN && relu)
    gemm_wmma_f32<true, false, false, false><<<grid, block, 0, st>>>(
        A, W, bias, C, M, N, K, lda, ldw, ldc, sA, sW, sC, alpha);
  else if (mode == NN && accum)
    gemm_wmma_f32<false, true, false, false><<<grid, block, 0, st>>>(
        A, W, bias, C, M, N, K, lda, ldw, ldc, sA, sW, sC, alpha);
  else if (mode == NT)
    gemm_wmma_f32<false, false, false, true><<<grid, block, 0, st>>>(
        A, W, bias, C, M, N, K, lda, ldw, ldc, sA, sW, sC, alpha);
  else  // TN
    gemm_wmma_f32<false, false, true, false><<<grid, block, 0, st>>>(
        A, W, bias, C, M, N, K, lda, ldw, ldc, sA, sW, sC, alpha);
}

extern "C" void kernel_launch(void* const* d_in, const int* in_sizes, int n_in,
                              void* d_out, int out_size, void* d_ws,
                              size_t ws_size, hipStream_t stream) {
  const float* node_features = (const float*)d_in[0];
  const float* edge_features = (const float*)d_in[1];
  const int* from_idx = (const int*)d_in[2];
  const int* to_idx = (const int*)d_in[3];
  const int* edge_indices = (const int*)d_in[4];
  const int* edge_counts = (const int*)d_in[5];
  const float* enc_Wn = (const float*)d_in[6];
  const float* enc_bn = (const float*)d_in[7];
  const float* enc_We = (const float*)d_in[8];
  const float* enc_be = (const float*)d_in[9];
  const float* comb_W1 = (const float*)d_in[10];
  const float* comb_b1 = (const float*)d_in[11];
  const float* comb_W2 = (const float*)d_in[12];
  const float* comb_b2 = (const float*)d_in[13];
  const float* msg_W1 = (const float*)d_in[14];
  const float* msg_b1 = (const float*)d_in[15];
  const float* msg_W2 = (const float*)d_in[16];
  const float* msg_b2 = (const float*)d_in[17];
  const float* rmsg_W1 = (const float*)d_in[18];
  const float* rmsg_b1 = (const float*)d_in[19];
  const float* rmsg_W2 = (const float*)d_in[20];
  const float* rmsg_b2 = (const float*)d_in[21];
  const float* upd_W1 = (const float*)d_in[22];
  const float* upd_b1 = (const float*)d_in[23];
  const float* upd_W2 = (const float*)d_in[24];
  const float* upd_b2 = (const float*)d_in[25];
  const float* t1_W = (const float*)d_in[26];
  const float* t1_b = (const float*)d_in[27];
  const float* t2_W = (const float*)d_in[28];
  const float* t2_b = (const float*)d_in[29];

  const long E = in_sizes[2];            // edges
  const long Nn = in_sizes[0] / 32;      // nodes (24576)
  const int G = in_sizes[5];             // 256 graphs
  const int PB = G / 2;                  // 128 pairs
  const long P = (long)G * 256;          // padded rows (65536)

  // ---- workspace layout (floats) ----
  float* w = (float*)d_ws;
  long off = 0;
  auto alloc = [&](long n) { float* p = w + off; off += n; return p; };
  float* h      = alloc(Nn * 64);
  float* e_enc  = alloc(E * 32);
  float* store  = alloc(E * 128);
  float* X1     = alloc(E * 160);  // also Xm
  float* T1     = alloc(E * 160);  // also Xr
  float* comb   = alloc(E * 32);   // 30 used, ld 32
  long hbuf = E * 128 > P * 64 ? E * 128 : P * 64;
  float* Hm     = alloc(hbuf);     // also Ht (P x 64)
  float* Hr     = alloc(hbuf);     // also Tt (P x 64)
  float* Mf     = alloc(E * 128);  // also messages
  float* Mb     = alloc(E * 128);
  float* agg    = alloc(Nn * 128);
  float* Xu     = alloc(Nn * 192);
  float* Hu     = alloc(Nn * 128);
  float* padded = alloc(P * 128);
  float* S      = alloc((long)PB * 256 * 256);
  float* inter  = alloc(P * 128);
  (void)ws_size; (void)n_in;

  const float TEMP_INV = 10.0f;  // 1/TEMP

  // encoders
  gemm(stream, NN, false, false, node_features, enc_Wn, enc_bn, h,
       (int)Nn, 64, 32, 32, 64, 64);
  gemm(stream, NN, false, false, edge_features, enc_We, enc_be, e_enc,
       (int)E, 32, 16, 16, 32, 32);
  zero_f32<<<nblk(E * 128), 256, 0, stream>>>(store, E * 128);
  zero_f32<<<nblk(P * 128), 256, 0, stream>>>(padded, P * 128);

  for (int it = 0; it < 3; ++it) {
    // combined = mlp2([e_enc, store])
    pack_comb<<<nblk(E * 160), 256, 0, stream>>>(e_enc, store, X1, E * 160);
    gemm(stream, NN, true, false, X1, comb_W1, comb_b1, T1,
         (int)E, 160, 160, 160, 160, 160);
    gemm(stream, NN, false, false, T1, comb_W2, comb_b2, comb,
         (int)E, 30, 160, 160, 30, 32);

    // forward / backward messages (pre-update h)
    pack_msg<<<nblk(E * 160), 256, 0, stream>>>(h, comb, from_idx, to_idx, X1,
                                                T1, E * 160);
    gemm(stream, NN, true, false, X1, msg_W1, msg_b1, Hm,
         (int)E, 128, 158, 160, 128, 128);
    gemm(stream, NN, false, false, Hm, msg_W2, msg_b2, Mf,
         (int)E, 128, 128, 128, 128, 128);
    gemm(stream, NN, true, false, T1, rmsg_W1, rmsg_b1, Hr,
         (int)E, 128, 158, 160, 128, 128);
    gemm(stream, NN, false, false, Hr, rmsg_W2, rmsg_b2, Mb,
         (int)E, 128, 128, 128, 128, 128);

    // agg = segment_sum(Mf,to) + segment_sum(Mb,from); h = mlp2([agg,h])
    zero_f32<<<nblk(Nn * 128), 256, 0, stream>>>(agg, Nn * 128);
    scatter_add_agg<<<nblk(E * 128), 256, 0, stream>>>(Mf, Mb, from_idx,
                                                       to_idx, agg, E * 128);
    pack_upd<<<nblk(Nn * 192), 256, 0, stream>>>(agg, h, Xu, Nn * 192);
    gemm(stream, NN, true, false, Xu, upd_W1, upd_b1, Hu,
         (int)Nn, 128, 192, 192, 128, 128);
    gemm(stream, NN, false, false, Hu, upd_W2, upd_b2, h,
         (int)Nn, 64, 128, 128, 64, 64);

    // messages = mlp_f + mlp_r (post-update h)
    pack_msg<<<nblk(E * 160), 256, 0, stream>>>(h, comb, from_idx, to_idx, X1,
                                                T1, E * 160);
    gemm(stream, NN, true, false, X1, msg_W1, msg_b1, Hm,
         (int)E, 128, 158, 160, 128, 128);
    gemm(stream, NN, false, false, Hm, msg_W2, msg_b2, Mf,
         (int)E, 128, 128, 128, 128, 128);
    gemm(stream, NN, true, false, T1, rmsg_W1, rmsg_b1, Hr,
         (int)E, 128, 158, 160, 128, 128);
    gemm(stream, NN, false, true, Hr, rmsg_W2, rmsg_b2, Mf,  // accumulate
         (int)E, 128, 128, 128, 128, 128);

    // scatter to padded [G*256,128]; untouched rows stay zero
    scatter_msgs<<<nblk(E * 128), 256, 0, stream>>>(Mf, edge_indices, padded,
                                                    E * 128);

    // tq/tc = mlp2(padded) masked  (Ht=Hm buf, Tt=Hr buf)
    gemm(stream, NN, true, false, padded, t1_W, t1_b, Hm,
         (int)P, 64, 128, 128, 64, 64);
    gemm(stream, NN, false, false, Hm, t2_W, t2_b, Hr,
         (int)P, 64, 64, 64, 64, 64);
    mask_rows<<<nblk(P * 64), 256, 0, stream>>>(Hr, edge_counts, P * 64);

    // S[b] = (Tq_b @ Tc_b^T) / TEMP   (batched NT)
    gemm(stream, NT, false, false, Hr, Hr + 256 * 64, nullptr, S,
         256, 256, 64, 64, 64, 256, PB, 2L * 256 * 64, 2L * 256 * 64,
         256L * 256, TEMP_INV);

    // sinkhorn (10 iters), then plan = exp(log_alpha) in place
    for (int si = 0; si < 10; ++si) {
      sinkhorn_row<<<PB * 256, 256, 0, stream>>>(S);
      sinkhorn_col<<<PB, 256, 0, stream>>>(S);
    }
    exp_inplace<<<nblk((long)PB * 256 * 256), 256, 0, stream>>>(
        S, (long)PB * 256 * 256);

    // q_from_c (even rows of inter) = plan @ c_store      (batched NN)
    gemm(stream, NN, false, false, S, padded + 256 * 128, nullptr, inter,
         256, 128, 256, 256, 128, 128, PB, 256L * 256, 2L * 256 * 128,
         2L * 256 * 128);
    // c_from_q (odd rows of inter) = plan^T @ q_store     (batched TN)
    gemm(stream, TN, false, false, S, padded, nullptr, inter + 256 * 128,
         256, 128, 256, 256, 128, 128, PB, 256L * 256, 2L * 256 * 128,
         2L * 256 * 128);

    if (it < 2)
      gather_store<<<nblk(E * 128), 256, 0, stream>>>(inter, edge_indices,
                                                      store, E * 128);
  }

  // scores[b] = -sum relu(q_store - q_from_c)
  final_scores<<<PB, 256, 0, stream>>>(padded, inter, (float*)d_out);
  (void)out_size;
}